// BlockSparseLinear_41128606826623
// MI455X (gfx1250) — compile-verified
//
#include <hip/hip_runtime.h>

// ---------------------------------------------------------------------------
// Block-sparse linear: y = x @ W^T + bias   (f32 in, f32 out)
// Compute-bound (477 FLOP/byte) -> native f32 WMMA (V_WMMA_F32_16X16X4_F32).
// Data movement: GLOBAL_LOAD_ASYNC_TO_LDS_B128 double-buffered staging
// (ASYNCcnt), one barrier per sparse block; next block's async loads and its
// packed (block,col) CSR entry are both pipelined so nothing but the WMMAs
// sits on the steady-state critical path.
// ---------------------------------------------------------------------------

#define BLK        32
#define IN_FEAT    4096
#define OUT_FEAT   4096
#define NBR        (OUT_FEAT / BLK)   // 128 block rows
#define NBC        (IN_FEAT / BLK)    // 128 block cols
#define MTILE      128                // tokens per workgroup
#define PITCH      36                 // 32 + 4 pad: 16B-aligned rows, conflict-free

typedef float v2f __attribute__((ext_vector_type(2)));
typedef float v4f __attribute__((ext_vector_type(4)));
typedef float v8f __attribute__((ext_vector_type(8)));
typedef int   v4i __attribute__((vector_size(16)));   // matches builtin param type

#if __has_builtin(__builtin_amdgcn_global_load_async_to_lds_b128)
#define HAVE_ASYNC 1
#else
#define HAVE_ASYNC 0
#endif

#define AS_GLOBAL __attribute__((address_space(1)))
#define AS_LOCAL  __attribute__((address_space(3)))

__device__ __forceinline__ void wait_async_zero() {
#if __has_builtin(__builtin_amdgcn_s_wait_asynccnt)
    __builtin_amdgcn_s_wait_asynccnt(0);
#else
    asm volatile("s_wait_asynccnt 0x0" ::: "memory");
#endif
}

#if HAVE_ASYNC
__device__ __forceinline__ void async_cp16(const float* gsrc, float* ldst) {
    __builtin_amdgcn_global_load_async_to_lds_b128(
        (AS_GLOBAL v4i*)gsrc, (AS_LOCAL v4i*)ldst, /*imm off*/0, /*cpol*/0);
}
#endif

// ---------------------------------------------------------------------------
// Kernel 1: deterministic CSR build. One wave32 per block-row scans block_rows
// in index order; ballot + prefix-popcount assigns slots. Entry packs the
// block id and its column: (b << 8) | c  -> GEMM needs ONE uniform load per
// block instead of a dependent csr->bcol chain.
// ---------------------------------------------------------------------------
__global__ __launch_bounds__(256) void bsl_build_csr(const int* __restrict__ brow,
                                                     const int* __restrict__ bcol,
                                                     int nnz,
                                                     int* __restrict__ csr,
                                                     int* __restrict__ cnt) {
    const int lane = threadIdx.x & 31;
    const int row  = (blockIdx.x * blockDim.x + threadIdx.x) >> 5;  // wave id
    if (row >= NBR) return;
    int off = 0;
    for (int base = 0; base < nnz; base += 32) {
        const int b = base + lane;
        bool m = false;
        if (b < nnz) m = (brow[b] == row);
        const unsigned mask = (unsigned)__ballot(m);          // wave32 low bits
        if (m) {
            const int pos = off + __popc(mask & ((1u << lane) - 1u));
            csr[row * NBC + pos] = (b << 8) | (bcol[b] & 0xff);
        }
        off += __popc(mask);
    }
    if (lane == 0) cnt[row] = off;
}

// ---------------------------------------------------------------------------
// Kernel 2: block-sparse GEMM via V_WMMA_F32_16X16X4_F32.
// grid = (n_tokens/128, NBR); 256 threads = 8 waves.
// Wave w: M-quadrant (w&3)*32 tokens, N-half (w>>2)*16 outs; 16 WMMAs/block.
// ---------------------------------------------------------------------------
__global__ __launch_bounds__(256) void bsl_gemm(const float* __restrict__ x,
                                                const float* __restrict__ wb,
                                                const float* __restrict__ bias,
                                                const int* __restrict__ csr,
                                                const int* __restrict__ cnt,
                                                float* __restrict__ out) {
#if HAVE_ASYNC
    __shared__ float xs[2][MTILE * PITCH];   // 2 x 18432 B
    __shared__ float wsh[2][BLK * PITCH];    // 2 x  4608 B   (total 45 KB)
#else
    __shared__ float xs[1][MTILE * PITCH];
    __shared__ float wsh[1][BLK * PITCH];
#endif

    const int tid  = threadIdx.x;
    const int lane = tid & 31;
    const int wid  = tid >> 5;            // 0..7
    const int m0   = (wid & 3) * 32;      // token sub-base within tile
    const int nb   = (wid >> 2) * 16;     // out-feature half within block row

    const int t0 = blockIdx.x * MTILE;    // token tile base
    const int r  = blockIdx.y;            // block row (32 out features)

    const int count = cnt[r];
    const int* rowcsr = csr + r * NBC;

    // Per-lane LDS fragment offsets (A = x rows; B[k][n] = block[n][k], i.e.
    // the row-major block read as its own transpose -> W^T for free).
    const int l15  = lane & 15;
    const int khi  = (lane >> 4) * 2;                 // K sub-offset: 0 or 2
    const int aoff0 = (m0 + l15) * PITCH + khi;       // tokens m0..m0+15
    const int aoff1 = aoff0 + 16 * PITCH;             // tokens m0+16..m0+31
    const int boff  = (nb + l15) * PITCH + khi;

    // Per-thread staging coordinates (1 x 16B for W, 4 x 16B for x).
    const int wn  = tid >> 3;             // weight row 0..31
    const int wk4 = (tid & 7) << 2;       // weight col chunk

    v8f acc0 = {0.f, 0.f, 0.f, 0.f, 0.f, 0.f, 0.f, 0.f};
    v8f acc1 = {0.f, 0.f, 0.f, 0.f, 0.f, 0.f, 0.f, 0.f};

#if HAVE_ASYNC
    // ---- CDNA5 path: async global->LDS staging, double buffered, with the
    //      CSR entry stream pipelined one block ahead. ----
    int e1 = 0;
    if (count > 0) {
        const int e0 = __builtin_amdgcn_readfirstlane(rowcsr[0]);
        if (count > 1) e1 = __builtin_amdgcn_readfirstlane(rowcsr[1]);
        const int b = e0 >> 8, c = e0 & 0xff;
        async_cp16(wb + (size_t)b * (BLK * BLK) + tid * 4, &wsh[0][wn * PITCH + wk4]);
        #pragma unroll
        for (int j = 0; j < 4; ++j) {
            const int e = tid + j * 256, t = e >> 3, k4 = (e & 7) << 2;
            async_cp16(x + (size_t)(t0 + t) * IN_FEAT + c * BLK + k4,
                       &xs[0][t * PITCH + k4]);
        }
    }
    for (int i = 0; i < count; ++i) {
        wait_async_zero();        // my async loads for buffer i&1 landed
        __syncthreads();          // everyone's landed; prior reads of other buf done
        if (i + 1 < count) {      // issue next block into the other buffer
            const int b2 = e1 >> 8, c2 = e1 & 0xff;   // entry prefetched last iter
            const int nbuf = (i + 1) & 1;
            async_cp16(wb + (size_t)b2 * (BLK * BLK) + tid * 4,
                       &wsh[nbuf][wn * PITCH + wk4]);
            #pragma unroll
            for (int j = 0; j < 4; ++j) {
                const int e = tid + j * 256, t = e >> 3, k4 = (e & 7) << 2;
                async_cp16(x + (size_t)(t0 + t) * IN_FEAT + c2 * BLK + k4,
                           &xs[nbuf][t * PITCH + k4]);
            }
            // prefetch the entry for block i+2 (hides under compute + barrier)
            if (i + 2 < count) e1 = __builtin_amdgcn_readfirstlane(rowcsr[i + 2]);
        }
        const float* xb = &xs[i & 1][0];
        const float* wf = &wsh[i & 1][0];
        #pragma unroll
        for (int k0 = 0; k0 < BLK; k0 += 4) {
            const v2f a0 = *(const v2f*)(xb + aoff0 + k0);
            const v2f a1 = *(const v2f*)(xb + aoff1 + k0);
            const v2f bf = *(const v2f*)(wf + boff + k0);
            acc0 = __builtin_amdgcn_wmma_f32_16x16x4_f32(
                       false, a0, false, bf, (short)0, acc0, false, false);
            acc1 = __builtin_amdgcn_wmma_f32_16x16x4_f32(
                       false, a1, false, bf, (short)0, acc1, false, false);
        }
    }
#else
    // ---- fallback: register staging, two barriers per block ----
    for (int i = 0; i < count; ++i) {
        const int e0 = rowcsr[i];
        const int b = e0 >> 8, c = e0 & 0xff;
        *(v4f*)(&wsh[0][wn * PITCH + wk4]) =
            *(const v4f*)(wb + (size_t)b * (BLK * BLK) + tid * 4);
        #pragma unroll
        for (int j = 0; j < 4; ++j) {
            const int e = tid + j * 256, t = e >> 3, k4 = (e & 7) << 2;
            *(v4f*)(&xs[0][t * PITCH + k4]) =
                *(const v4f*)(x + (size_t)(t0 + t) * IN_FEAT + c * BLK + k4);
        }
        if (i + 1 < count) {
            const int e1f = rowcsr[i + 1];
            const int b2 = e1f >> 8, c2 = e1f & 0xff;
            __builtin_prefetch(wb + (size_t)b2 * (BLK * BLK) + tid * 4, 0, 0);
            __builtin_prefetch(x + (size_t)(t0 + (tid >> 1)) * IN_FEAT + c2 * BLK +
                                   (tid & 1) * 16, 0, 0);
        }
        __syncthreads();
        #pragma unroll
        for (int k0 = 0; k0 < BLK; k0 += 4) {
            const v2f a0 = *(const v2f*)(&xs[0][aoff0 + k0]);
            const v2f a1 = *(const v2f*)(&xs[0][aoff1 + k0]);
            const v2f bf = *(const v2f*)(&wsh[0][boff + k0]);
            acc0 = __builtin_amdgcn_wmma_f32_16x16x4_f32(
                       false, a0, false, bf, (short)0, acc0, false, false);
            acc1 = __builtin_amdgcn_wmma_f32_16x16x4_f32(
                       false, a1, false, bf, (short)0, acc1, false, false);
        }
        __syncthreads();
    }
#endif

    // ---- epilogue: bias + non-temporal store (out is write-once; keep the
    //      hot x/W working set resident in the 192MB L2). C layout: VGPR v ->
    //      M=v (lanes 0-15) / M=v+8 (lanes 16-31); N = lane&15. ----
    const int col = r * BLK + nb + l15;
    const float bv = bias[col];
    const int rowbase = t0 + m0 + ((lane >> 4) << 3);   // +8 for hi half-wave
    #pragma unroll
    for (int v = 0; v < 8; ++v) {
        __builtin_nontemporal_store(acc0[v] + bv,
            out + (size_t)(rowbase + v) * OUT_FEAT + col);
        __builtin_nontemporal_store(acc1[v] + bv,
            out + (size_t)(rowbase + v + 16) * OUT_FEAT + col);
    }
}

// ---------------------------------------------------------------------------
extern "C" void kernel_launch(void* const* d_in, const int* in_sizes, int n_in,
                              void* d_out, int out_size, void* d_ws, size_t ws_size,
                              hipStream_t stream) {
    const float* x    = (const float*)d_in[0];
    const float* wb   = (const float*)d_in[1];
    const float* bias = (const float*)d_in[2];
    const int*   brow = (const int*)d_in[3];
    const int*   bcol = (const int*)d_in[4];
    const int nnz      = in_sizes[3];
    const int n_tokens = in_sizes[0] / IN_FEAT;

    int* csr = (int*)d_ws;          // NBR*NBC ints (64 KB)
    int* cnt = csr + NBR * NBC;     // NBR ints

    // 128 waves (one per block row) = 16 blocks x 256 threads.
    bsl_build_csr<<<(NBR * 32 + 255) / 256, 256, 0, stream>>>(brow, bcol, nnz, csr, cnt);

    dim3 grid(n_tokens / MTILE, NBR);
    bsl_gemm<<<grid, 256, 0, stream>>>(x, wb, bias, csr, cnt, (float*)d_out);
}